// ShellConv_79937931313698
// MI455X (gfx1250) — compile-verified
//
#include <hip/hip_runtime.h>

// ---------------------------------------------------------------------------
// ShellConv lift stage for MI455X (gfx1250, wave32, WMMA f32 16x16x4)
// ---------------------------------------------------------------------------
#define BLD   16        // B*L
#define NPTS  2048      // N
#define MQ    1024      // M
#define KNN   32        // neighbors
#define C1    32
#define C2    64
#define ROWS  (BLD * MQ * KNN)   // 524288
#define BN_EPS 1e-5f

typedef __attribute__((ext_vector_type(2))) float v2f;
typedef __attribute__((ext_vector_type(8))) float v8f;

// --------------------------- K0: zero stats region -------------------------
__global__ void k0_init(float* __restrict__ stats) {
    int i = threadIdx.x;
    if (i < 192) stats[i] = 0.0f;   // [0:64) layer1 sum/sumsq, [64:192) layer2
}

// --------- K1: per-query kNN (wave per query) + local coords + z1 stats ----
__global__ __launch_bounds__(256) void k1_topk(
    const float* __restrict__ points,    // [BLD, N, 3]
    const float* __restrict__ queries,   // [BLD, M, 3]
    const float* __restrict__ W1,        // [3,32]
    const float* __restrict__ b1,        // [32]
    float* __restrict__ localPad,        // [ROWS, 4]  (x,y,z,1)
    float* __restrict__ stats1)          // [64] sum/sumsq of z1
{
    __shared__ float4 pts[NPTS];
    __shared__ float  w1l[96];
    __shared__ float  b1l[32];
    __shared__ float  acc[64];

    const int tid  = threadIdx.x;
    const int lane = tid & 31;
    const int wave = tid >> 5;
    const int bl   = blockIdx.x >> 7;        // / (M/8)
    const int m    = (blockIdx.x & 127) * 8 + wave;

    const float* pbase = points + (size_t)bl * NPTS * 3;
    for (int i = tid; i < NPTS; i += 256)
        pts[i] = make_float4(pbase[i*3+0], pbase[i*3+1], pbase[i*3+2], 0.f);
    for (int i = tid; i < 96; i += 256) w1l[i] = W1[i];
    if (tid < 32) b1l[tid] = b1[tid];
    if (tid < 64) acc[tid] = 0.f;
    __syncthreads();

    const float* qp = queries + ((size_t)bl * MQ + m) * 3;
    const float qx = qp[0], qy = qp[1], qz = qp[2];

    // each lane owns 64 contiguous candidates in registers
    float d[64];
    #pragma unroll
    for (int j = 0; j < 64; ++j) {
        float4 p = pts[lane * 64 + j];
        float dx = qx - p.x, dy = qy - p.y, dz = qz - p.z;
        d[j] = dx*dx + dy*dy + dz*dz;
    }

    // 32 rounds of exact argmin extraction (tie-break: lower index)
    int sel = 0;
    for (int r = 0; r < KNN; ++r) {
        float lmin = 3.4e38f; int lidx = 0;
        #pragma unroll
        for (int j = 0; j < 64; ++j)
            if (d[j] < lmin) { lmin = d[j]; lidx = j; }
        int gidx = (lane << 6) | lidx;
        #pragma unroll
        for (int off = 16; off > 0; off >>= 1) {
            float ov = __shfl_xor(lmin, off, 32);
            int   oi = __shfl_xor(gidx, off, 32);
            if (ov < lmin || (ov == lmin && oi < gidx)) { lmin = ov; gidx = oi; }
        }
        if (lane == r) sel = gidx;               // round r => r-th nearest
        if ((gidx >> 6) == lane) {               // winner lane invalidates
            int wj = gidx & 63;
            #pragma unroll
            for (int j = 0; j < 64; ++j)
                if (j == wj) d[j] = 3.4e38f;
        }
    }

    const float4 p = pts[sel];
    const float lx = qx - p.x, ly = qy - p.y, lz = qz - p.z;
    const size_t row = (size_t)(bl * MQ + m) * KNN + lane;
    *(float4*)(localPad + row * 4) = make_float4(lx, ly, lz, 1.0f);

    // z1 = local @ W1 + b1  -> per-channel sum / sumsq (wave butterfly reduce)
    #pragma unroll
    for (int c = 0; c < C1; ++c) {
        float z = b1l[c] + lx * w1l[c] + ly * w1l[32 + c] + lz * w1l[64 + c];
        float zs = z * z;
        #pragma unroll
        for (int off = 16; off > 0; off >>= 1) {
            z  += __shfl_xor(z,  off, 32);
            zs += __shfl_xor(zs, off, 32);
        }
        if (lane == 0) { atomicAdd(&acc[c], z); atomicAdd(&acc[32 + c], zs); }
    }
    __syncthreads();
    if (tid < 64) atomicAdd(&stats1[tid], acc[tid]);
}

// ------------------- K2/K4: finalize BN affine parameters ------------------
__global__ void k_finalize(const float* __restrict__ stats,   // sum[C], sumsq[C]
                           const float* __restrict__ gamma,
                           const float* __restrict__ beta,
                           float* __restrict__ outParams,      // a[C], sh[C]
                           int C, float invCount)
{
    int c = threadIdx.x;
    if (c < C) {
        float mean = stats[c] * invCount;
        float var  = stats[C + c] * invCount - mean * mean;
        float a    = gamma[c] * rsqrtf(var + BN_EPS);
        outParams[c]     = a;
        outParams[C + c] = beta[c] - mean * a;
    }
}

// ----- K3: WMMA GEMM1 -> BN1/ReLU -> WMMA GEMM2 -> z2 + layer-2 stats ------
__global__ __launch_bounds__(256) void k3_gemm(
    const float* __restrict__ localPad,  // [ROWS,4]
    const float* __restrict__ W1,        // [3,32]
    const float* __restrict__ b1,        // [32]
    const float* __restrict__ W2,        // [32,64]
    const float* __restrict__ b2,        // [64]
    const float* __restrict__ params1,   // a1[32], sh1[32]
    float* __restrict__ out,             // [ROWS,64] (z2, pre-BN2)
    float* __restrict__ stats2)          // [128]
{
    __shared__ float w2l[2048];          // W2 staged
    __shared__ float hbuf[8 * 512];      // per-wave 16x32 h1 tile
    __shared__ float acc[128];

    const int tid  = threadIdx.x;
    const int lane = tid & 31;
    const int wave = tid >> 5;
    for (int i = tid; i < 2048; i += 256) w2l[i] = W2[i];
    if (tid < 128) acc[tid] = 0.f;
    __syncthreads();

    const int tile    = blockIdx.x * 8 + wave;
    const int rowBase = tile * 16;
    const int hi      = (lane & 16) ? 1 : 0;   // upper half-wave
    const int l15     = lane & 15;

    // A tile (16x4): lanes 0-15 hold K=0,1; lanes 16-31 hold K=2,3
    const float* ap = localPad + (size_t)(rowBase + l15) * 4 + hi * 2;
    const v2f aA = { ap[0], ap[1] };

    // GEMM1: z1 = [x y z 1] x [W1_0; W1_1; W1_2; b1], then BN1 + ReLU -> LDS
    float* hrow = &hbuf[wave * 512];
    #pragma unroll
    for (int t = 0; t < 2; ++t) {
        const int c1 = t * 16 + l15;
        v2f bB;
        bB[0] = hi ? W1[64 + c1] : W1[c1];        // row2 : row0
        bB[1] = hi ? b1[c1]      : W1[32 + c1];   // row3(=b1) : row1
        v8f cz = { 0.f, 0.f, 0.f, 0.f, 0.f, 0.f, 0.f, 0.f };
        cz = __builtin_amdgcn_wmma_f32_16x16x4_f32(
                 false, aA, false, bB, (short)0, cz, false, false);
        const float a  = params1[c1];
        const float sh = params1[32 + c1];
        #pragma unroll
        for (int j = 0; j < 8; ++j) {
            float h = cz[j] * a + sh;
            h = h > 0.f ? h : 0.f;
            hrow[(j + hi * 8) * 32 + c1] = h;     // C-layout -> row-major LDS
        }
    }
    __syncthreads();   // LDS re-layout fence (uniform across all 8 waves)

    // A slices for GEMM2 (K=32 -> 8 slices of 4)
    v2f asl[8];
    #pragma unroll
    for (int s = 0; s < 8; ++s) {
        const float* hp = &hrow[l15 * 32 + 4 * s + hi * 2];
        asl[s] = (v2f){ hp[0], hp[1] };
    }

    // GEMM2: z2 = h1 @ W2 + b2 (bias via C init); 4 n-tiles x 8 chained WMMAs
    #pragma unroll
    for (int u = 0; u < 4; ++u) {
        const int cc = u * 16 + l15;
        const float bb = b2[cc];
        v8f c8 = { bb, bb, bb, bb, bb, bb, bb, bb };
        #pragma unroll
        for (int s = 0; s < 8; ++s) {
            const int k0 = 4 * s + hi * 2;
            const v2f bv = { w2l[k0 * 64 + cc], w2l[(k0 + 1) * 64 + cc] };
            c8 = __builtin_amdgcn_wmma_f32_16x16x4_f32(
                     false, asl[s], false, bv, (short)0, c8, false, false);
        }
        // store z2 + accumulate column stats
        float psum = 0.f, psq = 0.f;
        #pragma unroll
        for (int j = 0; j < 8; ++j) {
            const float v = c8[j];
            out[(size_t)(rowBase + j + hi * 8) * 64 + cc] = v;
            psum += v; psq += v * v;
        }
        psum += __shfl_xor(psum, 16, 32);   // combine the two half-waves
        psq  += __shfl_xor(psq,  16, 32);
        if (!hi) { atomicAdd(&acc[cc], psum); atomicAdd(&acc[64 + cc], psq); }
    }
    __syncthreads();
    if (tid < 128) atomicAdd(&stats2[tid], acc[tid]);
}

// ----------------- K5: in-place BN2 + ReLU over d_out (float4) -------------
__global__ __launch_bounds__(256) void k5_bnrelu(
    float* __restrict__ out, const float* __restrict__ params2)
{
    const size_t i4   = (size_t)blockIdx.x * 256 + threadIdx.x;
    const size_t base = i4 * 4;
    const int c = (int)(base & 63);
    float4 v = *(float4*)(out + base);
    const float a0 = params2[c],      a1 = params2[c + 1];
    const float a2 = params2[c + 2],  a3 = params2[c + 3];
    const float s0 = params2[64 + c],     s1 = params2[64 + c + 1];
    const float s2 = params2[64 + c + 2], s3 = params2[64 + c + 3];
    v.x = fmaxf(v.x * a0 + s0, 0.f);
    v.y = fmaxf(v.y * a1 + s1, 0.f);
    v.z = fmaxf(v.z * a2 + s2, 0.f);
    v.w = fmaxf(v.w * a3 + s3, 0.f);
    *(float4*)(out + base) = v;
}

// ---------------------------------------------------------------------------
extern "C" void kernel_launch(void* const* d_in, const int* in_sizes, int n_in,
                              void* d_out, int out_size, void* d_ws, size_t ws_size,
                              hipStream_t stream) {
    (void)in_sizes; (void)n_in; (void)out_size; (void)ws_size;
    const float* points  = (const float*)d_in[0];
    const float* queries = (const float*)d_in[1];
    // d_in[2] = feat_prev (unused by reference)
    const float* W1     = (const float*)d_in[3];
    const float* b1     = (const float*)d_in[4];
    const float* gamma1 = (const float*)d_in[5];
    const float* beta1  = (const float*)d_in[6];
    const float* W2     = (const float*)d_in[7];
    const float* b2     = (const float*)d_in[8];
    const float* gamma2 = (const float*)d_in[9];
    const float* beta2  = (const float*)d_in[10];
    float* out = (float*)d_out;
    float* ws  = (float*)d_ws;

    float* localPad = ws;                          // ROWS*4 floats (8 MB)
    float* stats    = ws + (size_t)ROWS * 4;       // 192 floats
    float* params   = stats + 192;                 // 192 floats

    const float invCnt = 1.0f / (float)ROWS;

    k0_init   <<<1,     256, 0, stream>>>(stats);
    k1_topk   <<<BLD * (MQ / 8), 256, 0, stream>>>(points, queries, W1, b1,
                                                   localPad, stats);
    k_finalize<<<1,      64, 0, stream>>>(stats,      gamma1, beta1, params,      C1, invCnt);
    k3_gemm   <<<ROWS / 16 / 8, 256, 0, stream>>>(localPad, W1, b1, W2, b2,
                                                  params, out, stats + 64);
    k_finalize<<<1,      64, 0, stream>>>(stats + 64, gamma2, beta2, params + 64, C2, invCnt);
    k5_bnrelu <<<(ROWS * C2) / 4 / 256, 256, 0, stream>>>(out, params + 64);
}